// lossFunc_76665166233883
// MI455X (gfx1250) — compile-verified
//
#include <hip/hip_runtime.h>
#include <stdint.h>

// Problem geometry (fixed by the reference).
#define NB  512
#define NL  200
#define NH  2048
#define LM1 (NL - 1)      // 199 elements per batch row
#define NME (LM1 * 3)     // 597 int64s of `me` staged per block

// gfx1250 async global->LDS path (ASYNCcnt-tracked), guarded so the kernel
// still compiles if the toolchain doesn't declare these builtins.
#if defined(__has_builtin)
#if __has_builtin(__builtin_amdgcn_global_load_async_to_lds_b64) && \
    __has_builtin(__builtin_amdgcn_s_wait_asynccnt)
#define USE_ASYNC_LDS 1
#endif
#endif
#ifndef USE_ASYNC_LDS
#define USE_ASYNC_LDS 0
#endif

// Builtin signature (from hipcc diagnostic): param 0 is a pointer to a
// 2-int vector in the global (AS1 / __device__) address space, non-const.
typedef int v2i __attribute__((ext_vector_type(2)));
typedef __attribute__((address_space(1))) v2i* gv2i_p;  // global b64 chunk
typedef __attribute__((address_space(3))) v2i* lv2i_p;  // LDS b64 chunk

// One block per batch row b. 256 threads = 8 wave32s (good latency hiding for
// the scattered pred gather). Thread t handles l = t for t < 199.
__global__ __launch_bounds__(256) void bce_gather_kernel(
    const float*   __restrict__ pred,        // (B, L, H) f32
    const int64_t* __restrict__ me,          // (B, L, 3) i64
    float*         __restrict__ out,         // [0]=loss, [1..]=p flat, then a flat
    float*         __restrict__ batch_loss)  // d_ws: per-batch loss, NB floats
{
    const int b    = blockIdx.x;
    const int t    = threadIdx.x;
    const int lane = t & 31;                 // wave32 lane
    const int wv   = t >> 5;                 // wave index 0..7

    __shared__ int64_t s_me[NME];            // staged me[b, 1:, :]
    __shared__ float   s_sum[8];
    __shared__ int     s_max[8];

    // --- Stage the strided (24B-stride) me row as one coalesced LDS copy ---
    const int64_t* src = me + ((int64_t)b * NL + 1) * 3;
#if USE_ASYNC_LDS
    // global_load_async_to_lds_b64: DMA each 8B element straight into LDS
    // without transiting VGPRs; tracked by ASYNCcnt.
    for (int i = t; i < NME; i += 256) {
        __builtin_amdgcn_global_load_async_to_lds_b64(
            (gv2i_p)(int64_t*)(src + i), (lv2i_p)&s_me[i],
            /*imm offset*/0, /*cpol*/0);
    }
    __builtin_amdgcn_s_wait_asynccnt(0);     // this wave's async loads done
#else
    for (int i = t; i < NME; i += 256) s_me[i] = src[i];
#endif
    __syncthreads();                         // LDS visible block-wide

    // --- Gather p, emit p/a outputs, record validity candidate ---
    const bool active = (t < LM1);
    float p = 0.0f, a = 0.0f;
    int   myMax = -1;                        // max l with q[l] > 0 (suffix-any mask)
    if (active) {
        const int64_t q = s_me[t * 3 + 0];
        a = (float)s_me[t * 3 + 2];
        int idx = (int)q - 1;                // clamp like jnp.take_along_axis("clip")
        idx = idx < 0 ? 0 : (idx > NH - 1 ? NH - 1 : idx);
        p = pred[((int64_t)b * NL + t) * (int64_t)NH + idx];
        out[1 + (int64_t)b * LM1 + t]                          = p;
        out[1 + (int64_t)NB * LM1 + (int64_t)b * LM1 + t]      = a;
        if (q > 0) myMax = t;
    }

    // --- Block max-reduction (wave32 shuffles + tiny LDS pass) ---
    #pragma unroll
    for (int off = 16; off > 0; off >>= 1)
        myMax = max(myMax, __shfl_down(myMax, off, 32));
    if (lane == 0) s_max[wv] = myMax;
    __syncthreads();
    int maxValid = -1;
    #pragma unroll
    for (int i = 0; i < 8; ++i) maxValid = max(maxValid, s_max[i]);

    // --- Masked BCE ---
    float bce = 0.0f;
    if (active && t <= maxValid) {
        // -(a*log(p) + (1-a)*log1p(-p)), matching the reference in f32
        bce = -(a * logf(p) + (1.0f - a) * log1pf(-p));
    }

    // --- Block sum-reduction (deterministic: fixed tree order) ---
    #pragma unroll
    for (int off = 16; off > 0; off >>= 1)
        bce += __shfl_down(bce, off, 32);
    if (lane == 0) s_sum[wv] = bce;
    __syncthreads();
    if (t == 0) {
        float s = 0.0f;
        #pragma unroll
        for (int i = 0; i < 8; ++i) s += s_sum[i];
        const float denom = fmaxf((float)(maxValid + 1), 1.0f);
        batch_loss[b] = s / denom;
    }
}

// Deterministic 512 -> 1 reduction (single block, fixed LDS tree order).
// No float atomics, so graph replays are bit-identical.
__global__ __launch_bounds__(256) void final_reduce_kernel(
    const float* __restrict__ batch_loss,
    float*       __restrict__ out)
{
    __shared__ float s[256];
    const int t = threadIdx.x;
    s[t] = batch_loss[t] + batch_loss[t + 256];
    __syncthreads();
    #pragma unroll
    for (int stride = 128; stride > 0; stride >>= 1) {
        if (t < stride) s[t] += s[t + stride];
        __syncthreads();
    }
    if (t == 0) out[0] = s[0];
}

extern "C" void kernel_launch(void* const* d_in, const int* in_sizes, int n_in,
                              void* d_out, int out_size, void* d_ws, size_t ws_size,
                              hipStream_t stream)
{
    (void)in_sizes; (void)n_in; (void)out_size; (void)ws_size;
    const float*   pred = (const float*)d_in[0];      // f32 (B, L, H)
    const int64_t* me   = (const int64_t*)d_in[1];    // i64 (B, L, 3)
    float* out = (float*)d_out;
    float* ws  = (float*)d_ws;                        // NB floats of scratch

    bce_gather_kernel<<<NB, 256, 0, stream>>>(pred, me, out, ws);
    final_reduce_kernel<<<1, 256, 0, stream>>>(ws, out);
}